// MultiHeadGATLayer_70291434766877
// MI455X (gfx1250) — compile-verified
//
#include <hip/hip_runtime.h>
#include <math.h>

#define N_NODES 50000
#define N_EDGES 1600000
#define IN_DIM 128
#define HID 32
#define HEADS 4
#define OUT_DIM (HEADS * HID)   // 128
#define NEG_SLOPE 0.2f

typedef float v2f __attribute__((ext_vector_type(2)));
typedef float v8f __attribute__((ext_vector_type(8)));

// ---------------------------------------------------------------------------
// Stage A: QKV projection via f32 WMMA (V_WMMA_F32_16X16X4_F32).
// One wave -> one 16x16 output tile of one of {Q,K,V}.
// grid = (N_NODES/16, 24); blockDim = 32 (one wave).
//   blockIdx.y: mat = y/8 (0=Q,1=K,2=V), column tile = y%8 (16 cols each).
// A-layout (16x4 f32, 2 VGPRs): lane L holds A[L%16][2*(L/16)+{0,1}]  -> float2 load
// B-layout (4x16):              lane L holds B[2*(L/16)+{0,1}][L%16]
// D-layout (16x16 f32, 8 VGPRs): vgpr r -> row r + 8*(L/16), col L%16
// ---------------------------------------------------------------------------
__global__ __launch_bounds__(32)
void qkv_proj_wmma(const float* __restrict__ h,
                   const float* __restrict__ Wq,
                   const float* __restrict__ Wk,
                   const float* __restrict__ Wv,
                   float* __restrict__ Q,
                   float* __restrict__ K,
                   float* __restrict__ V) {
  const int lane = threadIdx.x & 31;
  const int m16  = lane & 15;
  const int half = lane >> 4;               // 0: lanes 0-15, 1: lanes 16-31
  const int rowBase = blockIdx.x * 16;
  const int mat  = blockIdx.y >> 3;
  const int col0 = (blockIdx.y & 7) * 16;

  const float* W   = (mat == 0) ? Wq : (mat == 1) ? Wk : Wv;
  float*       Out = (mat == 0) ? Q  : (mat == 1) ? K  : V;

  const int j    = col0 + m16;              // output column 0..127
  const int head = j >> 5;
  const int dcol = j & 31;
  // W[head][k][dcol]: stride HID over k
  const float* Wh   = W + (size_t)head * (IN_DIM * HID) + dcol;
  const float* Arow = h + (size_t)(rowBase + m16) * IN_DIM + half * 2;

  v8f c = {};
  #pragma unroll 4
  for (int kb = 0; kb < IN_DIM; kb += 4) {
    v2f a;
    a.x = Arow[kb + 0];                     // K = kb + 2*half
    a.y = Arow[kb + 1];                     // K = kb + 2*half + 1
    v2f b;
    b.x = Wh[(size_t)(kb + half * 2 + 0) * HID];
    b.y = Wh[(size_t)(kb + half * 2 + 1) * HID];
    c = __builtin_amdgcn_wmma_f32_16x16x4_f32(false, a, false, b,
                                              (short)0, c, false, false);
  }
  #pragma unroll
  for (int r = 0; r < 8; ++r) {
    Out[(size_t)(rowBase + r + half * 8) * OUT_DIM + col0 + m16] = c[r];
  }
}

// ---------------------------------------------------------------------------
// Init: segmax = -inf, denom = 0, out = 0
// ---------------------------------------------------------------------------
__global__ __launch_bounds__(256)
void init_buffers(float* __restrict__ segmax, float* __restrict__ denom,
                  float* __restrict__ out) {
  int i = blockIdx.x * blockDim.x + threadIdx.x;
  if (i < N_NODES * HEADS) {
    segmax[i] = -INFINITY;
    denom[i]  = 0.0f;
  }
  if (i < N_NODES * OUT_DIM) out[i] = 0.0f;
}

// Monotone float max via integer atomics (works for mixed signs, init = -inf).
__device__ __forceinline__ void atomicMaxFloat(float* addr, float val) {
  if (val >= 0.0f) atomicMax((int*)addr, __float_as_int(val));
  else             atomicMin((unsigned int*)addr, __float_as_uint(val));
}

// ---------------------------------------------------------------------------
// Stage B1: e = leaky_relu(<q_src, k_dst>) per (edge, head); seg-max into dst.
// Q/K are L2-resident (25.6 MB each on a 192 MB L2).
// ---------------------------------------------------------------------------
__global__ __launch_bounds__(256)
void edge_scores(const float* __restrict__ Q, const float* __restrict__ K,
                 const int* __restrict__ src, const int* __restrict__ dst,
                 float* __restrict__ e, float* __restrict__ segmax) {
  int t = blockIdx.x * blockDim.x + threadIdx.x;
  if (t >= N_EDGES * HEADS) return;
  const int edge = t >> 2, hd = t & 3;
  const int s = src[edge], d = dst[edge];
  const float4* qp = (const float4*)(Q + (size_t)s * OUT_DIM + hd * HID);
  const float4* kp = (const float4*)(K + (size_t)d * OUT_DIM + hd * HID);
  float acc = 0.0f;
  #pragma unroll
  for (int i = 0; i < 8; ++i) {
    float4 a = qp[i], b = kp[i];
    acc += a.x * b.x + a.y * b.y + a.z * b.z + a.w * b.w;
  }
  const float ev = acc > 0.0f ? acc : NEG_SLOPE * acc;
  e[t] = ev;
  atomicMaxFloat(&segmax[d * HEADS + hd], ev);
}

// ---------------------------------------------------------------------------
// Stage B2: ex = exp(e - segmax[dst]); seg-sum into denom[dst].
// ---------------------------------------------------------------------------
__global__ __launch_bounds__(256)
void edge_exp(float* __restrict__ e, const int* __restrict__ dst,
              const float* __restrict__ segmax, float* __restrict__ denom) {
  int t = blockIdx.x * blockDim.x + threadIdx.x;
  if (t >= N_EDGES * HEADS) return;
  const int edge = t >> 2, hd = t & 3;
  const int d = dst[edge];
  const float ex = __expf(e[t] - segmax[d * HEADS + hd]);
  e[t] = ex;                                   // reuse buffer for ex
  atomicAdd(&denom[d * HEADS + hd], ex);
}

// ---------------------------------------------------------------------------
// Stage B3: out[dst] += (ex/denom) * v[src]  (32 f32 atomics per (edge,head))
// ---------------------------------------------------------------------------
__global__ __launch_bounds__(256)
void aggregate(const float* __restrict__ ex, const float* __restrict__ denom,
               const float* __restrict__ V,
               const int* __restrict__ src, const int* __restrict__ dst,
               float* __restrict__ out) {
  int t = blockIdx.x * blockDim.x + threadIdx.x;
  if (t >= N_EDGES * HEADS) return;
  const int edge = t >> 2, hd = t & 3;
  const int s = src[edge], d = dst[edge];
  const float alpha = ex[t] / fmaxf(denom[d * HEADS + hd], 1e-9f);
  const float4* vp = (const float4*)(V + (size_t)s * OUT_DIM + hd * HID);
  float* op = out + (size_t)d * OUT_DIM + hd * HID;
  #pragma unroll
  for (int i = 0; i < 8; ++i) {
    float4 v4 = vp[i];
    atomicAdd(op + 4 * i + 0, alpha * v4.x);
    atomicAdd(op + 4 * i + 1, alpha * v4.y);
    atomicAdd(op + 4 * i + 2, alpha * v4.z);
    atomicAdd(op + 4 * i + 3, alpha * v4.w);
  }
}

// ---------------------------------------------------------------------------
extern "C" void kernel_launch(void* const* d_in, const int* in_sizes, int n_in,
                              void* d_out, int out_size, void* d_ws, size_t ws_size,
                              hipStream_t stream) {
  (void)in_sizes; (void)n_in; (void)out_size; (void)ws_size;
  const float* h   = (const float*)d_in[0];
  const int*   src = (const int*)  d_in[1];
  const int*   dst = (const int*)  d_in[2];
  const float* Wq  = (const float*)d_in[3];
  const float* Wk  = (const float*)d_in[4];
  const float* Wv  = (const float*)d_in[5];
  float* out = (float*)d_out;

  // Workspace layout (floats)
  float* ws = (float*)d_ws;
  const size_t NQ = (size_t)N_NODES * OUT_DIM;   // 6.4M
  float* Q      = ws;
  float* K      = Q + NQ;
  float* V      = K + NQ;
  float* ebuf   = V + NQ;                         // N_EDGES*HEADS
  float* segmax = ebuf + (size_t)N_EDGES * HEADS;
  float* denom  = segmax + (size_t)N_NODES * HEADS;

  // init out / segmax / denom
  {
    const int total = N_NODES * OUT_DIM;          // covers N_NODES*HEADS too
    init_buffers<<<(total + 255) / 256, 256, 0, stream>>>(segmax, denom, out);
  }
  // QKV projection: 3125 row tiles x (3 mats * 8 col tiles)
  {
    dim3 grid(N_NODES / 16, 24);
    qkv_proj_wmma<<<grid, 32, 0, stream>>>(h, Wq, Wk, Wv, Q, K, V);
  }
  const int ethreads = N_EDGES * HEADS;
  const int eblocks  = (ethreads + 255) / 256;
  edge_scores<<<eblocks, 256, 0, stream>>>(Q, K, src, dst, ebuf, segmax);
  edge_exp   <<<eblocks, 256, 0, stream>>>(ebuf, dst, segmax, denom);
  aggregate  <<<eblocks, 256, 0, stream>>>(ebuf, denom, V, src, dst, out);
}